// BatchSpecAugment_6743098654723
// MI455X (gfx1250) — compile-verified
//
#include <hip/hip_runtime.h>
#include <hip/hip_bf16.h>

// SpecAugment for MI455X (gfx1250).
// Memory-bound: 131MB in + 131MB out => ~11.2us at 23.3 TB/s. No FLOPs, so no
// WMMA; instead we use the CDNA5 async-to-LDS load path (ASYNCcnt) for the
// input stream and non-temporal b128 stores for the output stream.

#define USE_ASYNC_LDS 1

typedef __attribute__((ext_vector_type(4))) float v4f;

// ---------- JAX threefry2x32 (partitionable mode, JAX >= 0.5 default) ------
__device__ __forceinline__ unsigned rotl32(unsigned x, unsigned r) {
  return (x << r) | (x >> (32u - r));
}

__device__ __forceinline__ void threefry2x32(unsigned k0, unsigned k1,
                                             unsigned c0, unsigned c1,
                                             unsigned& o0, unsigned& o1) {
  unsigned ks0 = k0, ks1 = k1, ks2 = 0x1BD11BDAu ^ k0 ^ k1;
  unsigned x0 = c0 + ks0, x1 = c1 + ks1;
#define TF_GROUP(r0, r1, r2, r3, ka, kb, inc)                                  \
  x0 += x1; x1 = rotl32(x1, r0); x1 ^= x0;                                     \
  x0 += x1; x1 = rotl32(x1, r1); x1 ^= x0;                                     \
  x0 += x1; x1 = rotl32(x1, r2); x1 ^= x0;                                     \
  x0 += x1; x1 = rotl32(x1, r3); x1 ^= x0;                                     \
  x0 += ka; x1 += kb + inc;
  TF_GROUP(13u, 15u, 26u, 6u,  ks1, ks2, 1u)
  TF_GROUP(17u, 29u, 16u, 24u, ks2, ks0, 2u)
  TF_GROUP(13u, 15u, 26u, 6u,  ks0, ks1, 3u)
  TF_GROUP(17u, 29u, 16u, 24u, ks1, ks2, 4u)
  TF_GROUP(13u, 15u, 26u, 6u,  ks2, ks0, 5u)
#undef TF_GROUP
  o0 = x0; o1 = x1;
}

// 32-bit random_bits, partitionable threefry: element i -> xor of both lanes
__device__ __forceinline__ unsigned tf_bits32(unsigned k0, unsigned k1,
                                              unsigned i) {
  unsigned o0, o1;
  threefry2x32(k0, k1, 0u, i, o0, o1);
  return o0 ^ o1;
}

// jax.random.uniform f32 mapping: [0,1)
__device__ __forceinline__ float u01(unsigned bits) {
  return __uint_as_float((bits >> 9) | 0x3f800000u) - 1.0f;
}

// ---------------- per-batch mask parameters --------------------------------
struct Params {
  int ts0, te0, ts1, te1;   // time-mask intervals [ts, te)
  int fs0, fe0, fs1, fe1;   // freq-mask intervals [fs, fe)
  int vt;                   // valid length
  int pad0, pad1, pad2;     // pad to 48B (3 x b128 scalar loads)
};

__global__ void specaug_params_kernel(const int* __restrict__ lengths,
                                      Params* __restrict__ prm) {
  const int b = (int)threadIdx.x;
  if (b >= 64) return;

  // root key for jax.random.key(42) is (0, 42)
  // split(root, 4) fold-like: child c = threefry(root, 0, c)
  unsigned kfw0, kfw1, kfs0, kfs1, ktw0, ktw1, kts0, kts1;
  threefry2x32(0u, 42u, 0u, 0u, kfw0, kfw1);  // kf_w
  threefry2x32(0u, 42u, 0u, 1u, kfs0, kfs1);  // kf_s
  threefry2x32(0u, 42u, 0u, 2u, ktw0, ktw1);  // kt_w
  threefry2x32(0u, 42u, 0u, 3u, kts0, kts1);  // kt_s
  // randint(kf_w, ...) internally splits kf_w into (high-bits key, low-bits key)
  unsigned hk0, hk1, lk0, lk1;
  threefry2x32(kfw0, kfw1, 0u, 0u, hk0, hk1);
  threefry2x32(kfw0, kfw1, 0u, 1u, lk0, lk1);

  const int vt = lengths[b];

  // ---- frequency masks: fwidths = randint(0, 31), fstart = floor(u*(F-w+1))
  int fs[2], fe[2];
  const unsigned span = 31u;                       // fw_max + 1, fw_max=30
  unsigned mult = 65536u % span;                   // 2^(nbits/2) % span
  mult = (mult * mult) % span;
  for (int m = 0; m < 2; ++m) {
    const unsigned i = (unsigned)(b * 2 + m);
    const unsigned hb = tf_bits32(hk0, hk1, i) % span;
    const unsigned lb = tf_bits32(lk0, lk1, i) % span;
    const int fw = (int)((hb * mult + lb) % span);
    const float fu = u01(tf_bits32(kfs0, kfs1, i));
    const int fstart = (int)floorf(fu * (float)(128 - fw + 1));
    const bool fvalid = (fw > 0) && (fw < 128);
    fs[m] = fvalid ? fstart : 0;
    fe[m] = fvalid ? fstart + fw : 0;
  }

  // ---- time masks (bounds depend on per-sample valid length)
  int ts[2], te[2];
  const int twmax = vt < 40 ? vt : 40;             // min(TIME_MASK_PARAM, vt)
  for (int m = 0; m < 2; ++m) {
    const unsigned i = (unsigned)(b * 2 + m);
    const float tu = u01(tf_bits32(ktw0, ktw1, i));
    int tw = (int)floorf(tu * (float)(twmax + 1));
    tw = tw < twmax ? tw : twmax;
    const float su = u01(tf_bits32(kts0, kts1, i));
    int sr = vt - tw + 1;
    if (sr < 1) sr = 1;
    const int tstart = (int)floorf(su * (float)sr);
    const bool tvalid = (tw > 0) && (tw < vt) && (vt > 0);
    ts[m] = tvalid ? tstart : 0;
    te[m] = tvalid ? tstart + tw : 0;
  }

  Params p;
  p.ts0 = ts[0]; p.te0 = te[0]; p.ts1 = ts[1]; p.te1 = te[1];
  p.fs0 = fs[0]; p.fe0 = fe[0]; p.fs1 = fs[1]; p.fe1 = fe[1];
  p.vt = vt; p.pad0 = 0; p.pad1 = 0; p.pad2 = 0;
  prm[b] = p;
}

// ---------------- streaming mask-apply kernel ------------------------------
// One float4 (4 freq bins) per lane. 4000*32 = 128000 float4 per batch row;
// 128000/256 = 500 blocks per batch => `b` is uniform per block (scalar loads
// for Params, SALU index math).
__global__ __launch_bounds__(256) void specaug_apply_kernel(
    const v4f* __restrict__ mel, v4f* __restrict__ out,
    const Params* __restrict__ prm) {
  constexpr int F4_PER_B = 4000 * 32;        // 128000
  constexpr int BLOCKS_PER_B = F4_PER_B / 256;  // 500
  const int b = (int)blockIdx.x / BLOCKS_PER_B;
  const int within = ((int)blockIdx.x % BLOCKS_PER_B) * 256 + (int)threadIdx.x;
  const int t = within >> 5;                 // row (time index)
  const int f0 = (within & 31) << 2;         // first freq bin of this float4
  const int idx = b * F4_PER_B + within;     // < 8.192M, fits in int

  const Params p = prm[b];
  const bool tm = ((t >= p.ts0) & (t < p.te0)) | ((t >= p.ts1) & (t < p.te1));
  const bool inv = t < p.vt;

#if USE_ASYNC_LDS
  // CDNA5 async data path: HBM -> LDS without touching VGPRs (ASYNCcnt),
  // then ds_load_b128 back. Each lane owns its private 16B LDS slot, so no
  // barrier is needed, only s_wait_asynccnt for this wave.
  __shared__ v4f tile[256];
  {
    const unsigned lds = (unsigned)(size_t)(&tile[threadIdx.x]);
    const unsigned long long g = (unsigned long long)(size_t)(mel + idx);
    asm volatile("global_load_async_to_lds_b128 %0, %1, off"
                 :: "v"(lds), "v"(g) : "memory");
    asm volatile("s_wait_asynccnt 0" ::: "memory");
  }
  const v4f v = tile[threadIdx.x];
#else
  const v4f v = __builtin_nontemporal_load(mel + idx);
#endif

  v4f r;
#pragma unroll
  for (int j = 0; j < 4; ++j) {
    const int f = f0 + j;
    const bool fm =
        ((((f >= p.fs0) & (f < p.fe0)) | ((f >= p.fs1) & (f < p.fe1))) && inv);
    r[j] = (tm | fm) ? 0.0f : v[j];
  }
  // Output is streamed once and never re-read: non-temporal store keeps L2
  // clean (th:NT on global_store_b128).
  __builtin_nontemporal_store(r, out + idx);
}

extern "C" void kernel_launch(void* const* d_in, const int* in_sizes, int n_in,
                              void* d_out, int out_size, void* d_ws,
                              size_t ws_size, hipStream_t stream) {
  (void)in_sizes; (void)n_in; (void)out_size; (void)ws_size;
  const float* mel = (const float*)d_in[0];
  const int* lengths = (const int*)d_in[1];   // int64 in source demotes to i32 (no x64)
  float* out = (float*)d_out;
  Params* prm = (Params*)d_ws;                // 64 * 48B = 3KB scratch

  specaug_params_kernel<<<1, 64, 0, stream>>>(lengths, prm);

  constexpr int TOTAL_F4 = 64 * 4000 * 32;    // 8,192,000
  specaug_apply_kernel<<<TOTAL_F4 / 256, 256, 0, stream>>>(
      (const v4f*)mel, (v4f*)out, prm);
}